// WeightedAggEdge_67439576482329
// MI455X (gfx1250) — compile-verified
//
#include <hip/hip_runtime.h>
#include <hip/hip_bf16.h>
#include <math.h>

#define N_NODES   50000
#define N_EDGES   800000
#define NODE_DIM  128
#define EDGE_DIM  128
#define OUT_DIM   (NODE_DIM + EDGE_DIM)   // 256
#define TILE_ROWS 64                      // edges per tile (32 KB LDS per buffer)
#define N_TILES   ((N_EDGES + TILE_ROWS - 1) / TILE_ROWS)
#define SCATTER_BLOCKS 2500               // persistent blocks (~5 tiles each)

typedef __attribute__((ext_vector_type(4))) unsigned int u32x4;
typedef __attribute__((ext_vector_type(8))) int          i32x8;
typedef __attribute__((ext_vector_type(4))) int          i32x4;

#if __has_builtin(__builtin_amdgcn_tensor_load_to_lds)
#define HAVE_TDM 1
#else
#define HAVE_TDM 0
#endif

#if HAVE_TDM
// Issue one Tensor Data Mover load: rows x 128 f32 tile, global -> LDS.
__device__ __forceinline__ void tdm_load_tile(const float* gsrc,
                                              unsigned lds_byte_addr, int rows) {
  unsigned long long ga = (unsigned long long)(size_t)gsrc;
  // D# group 0: count=1(valid), lds_addr, 57b global_addr, type=2 ("image")
  u32x4 g0 = { 1u,
               lds_byte_addr,
               (unsigned int)ga,
               (unsigned int)((ga >> 32) & 0x1FFFFFFull) | (2u << 30) };
  // D# group 1: mask=0 (no cluster), data_size=2 (4B), tensor_dim0=128,
  // tensor_dim1=rows, tile_dim0=128, tile_dim1=rows, dim0_stride=128.
  i32x8 g1 = { (int)(2u << 16),
               (int)((unsigned)EDGE_DIM << 16),
               (int)((unsigned)rows << 16),
               (int)((unsigned)EDGE_DIM << 16),
               rows,
               EDGE_DIM,
               0, 0 };
  i32x4 z4 = {0, 0, 0, 0};
  i32x8 z8 = {0, 0, 0, 0, 0, 0, 0, 0};
  __builtin_amdgcn_tensor_load_to_lds(g0, g1, z4, z4, z8, 0);
}
#endif

// ---------------- K1a: zero deg + bincount workspace -----------------------
__global__ void wa_zero_i32(int* __restrict__ p, int n) {
  int i = blockIdx.x * blockDim.x + threadIdx.x;
  if (i < n) p[i] = 0;
}

// ---------------- K1b: zero the accumulation half of d_out -----------------
__global__ void wa_zero_accum(float* __restrict__ out) {
  int i = blockIdx.x * blockDim.x + threadIdx.x;
  if (i < N_NODES * EDGE_DIM) {
    int node = i >> 7;
    int col  = i & (EDGE_DIM - 1);
    out[(size_t)node * OUT_DIM + NODE_DIM + col] = 0.0f;
  }
}

// ---------------- K2: double-buffered TDM edge scatter ---------------------
// Persistent blocks; wave 0 keeps two TDM loads in flight (TENSORcnt<=2) so
// the DMA for tile t+1 streams from HBM while 8 waves drain tile t into
// global_atomic_add_f32 on the L2-resident accumulator (d_out[:,128:]).
__global__ void wa_edge_scatter(const float* __restrict__ e_lbl,
                                const int*   __restrict__ edge_dst,
                                float*       __restrict__ out,
                                int*         __restrict__ deg,
                                int n_tiles) {
  __shared__ float tile[2][TILE_ROWS * EDGE_DIM];
  const int wave   = threadIdx.x >> 5;
  const int lane   = threadIdx.x & 31;
  const int stride = gridDim.x;

#if HAVE_TDM
  if (threadIdx.x < 32) {                      // wave 0 is the DMA producer
    int t0 = blockIdx.x;
    if (t0 < n_tiles) {
      int s = t0 * TILE_ROWS;
      int r = min(TILE_ROWS, N_EDGES - s);
      tdm_load_tile(e_lbl + (size_t)s * EDGE_DIM,
                    (unsigned)(size_t)(void*)&tile[0][0], r);
    }
    int t1 = t0 + stride;
    if (t1 < n_tiles) {
      int s = t1 * TILE_ROWS;
      int r = min(TILE_ROWS, N_EDGES - s);
      tdm_load_tile(e_lbl + (size_t)s * EDGE_DIM,
                    (unsigned)(size_t)(void*)&tile[1][0], r);
    }
  }
#endif

  int buf = 0;
  for (int t = blockIdx.x; t < n_tiles; t += stride, buf ^= 1) {
    const int tile_start = t * TILE_ROWS;
    const int rows = min(TILE_ROWS, N_EDGES - tile_start);

#if HAVE_TDM
    if (threadIdx.x < 32) {
      // Outstanding here: {L(t), maybe L(t+stride)} -> wait for oldest.
      if (t + stride < n_tiles) __builtin_amdgcn_s_wait_tensorcnt(1);
      else                      __builtin_amdgcn_s_wait_tensorcnt(0);
    }
#else
    for (int i = threadIdx.x; i < rows * EDGE_DIM; i += blockDim.x)
      tile[buf][i] = e_lbl[(size_t)tile_start * EDGE_DIM + i];
#endif

    // in-degree counting (independent of the LDS tile)
    if ((int)threadIdx.x < rows)
      atomicAdd(&deg[edge_dst[tile_start + threadIdx.x]], 1);

    __syncthreads();   // tile[buf] ready for all waves

    // 8 waves x 8 rows; each atomic instruction covers 32 consecutive floats
    // (128B = 2 cachelines, 16 lanes/line) for maximal L2 atomic packing.
    for (int i = 0; i < 8; ++i) {
      const int r = wave * 8 + i;
      if (r >= rows) break;
      const int dst = edge_dst[tile_start + r];   // wave-uniform -> scalar load
      float* __restrict__ orow = out + (size_t)dst * OUT_DIM + NODE_DIM;
      const float* lrow = &tile[buf][r * EDGE_DIM];
#pragma unroll
      for (int k = 0; k < 4; ++k)
        atomicAdd(&orow[k * 32 + lane], lrow[k * 32 + lane]);
    }

    __syncthreads();   // everyone done reading tile[buf]; safe to refill

#if HAVE_TDM
    if (threadIdx.x < 32) {
      int tn = t + 2 * stride;                  // prefetch two tiles ahead
      if (tn < n_tiles) {
        int s = tn * TILE_ROWS;
        int r = min(TILE_ROWS, N_EDGES - s);
        tdm_load_tile(e_lbl + (size_t)s * EDGE_DIM,
                      (unsigned)(size_t)(void*)&tile[buf][0], r);
      }
    }
#endif
  }
}

// ---------------- K3: bincount of degrees ----------------------------------
__global__ void wa_bincount(const int* __restrict__ deg, int* __restrict__ counts) {
  int i = blockIdx.x * blockDim.x + threadIdx.x;
  if (i < N_NODES) atomicAdd(&counts[deg[i]], 1);
}

// ---------------- K4: normalize + elu + zero-degree mask -------------------
__global__ void wa_finalize(const float* __restrict__ h,
                            const int*   __restrict__ deg,
                            const int*   __restrict__ counts,
                            float*       __restrict__ out) {
  int i = blockIdx.x * blockDim.x + threadIdx.x;
  if (i >= N_NODES * OUT_DIM) return;
  const int node = i >> 8;            // OUT_DIM = 256
  const int col  = i & (OUT_DIM - 1);
  const int d    = deg[node];
  float v = 0.0f;                     // deg==0 rows stay zero (reference where())
  if (d > 0) {
    float x;
    if (col < NODE_DIM) {
      x = h[(size_t)node * NODE_DIM + col];
    } else {
      x = out[i] / (float)counts[d];  // accum_sum / bucket_size
    }
    v = (x > 0.0f) ? x : expm1f(x);   // elu, alpha=1
  }
  out[i] = v;
}

// ---------------- launch ---------------------------------------------------
extern "C" void kernel_launch(void* const* d_in, const int* in_sizes, int n_in,
                              void* d_out, int out_size, void* d_ws, size_t ws_size,
                              hipStream_t stream) {
  (void)in_sizes; (void)n_in; (void)out_size; (void)ws_size;
  const float* h        = (const float*)d_in[0];
  const float* e_lbl    = (const float*)d_in[1];
  const int*   edge_dst = (const int*)d_in[2];
  float* out  = (float*)d_out;
  int* deg    = (int*)d_ws;                 // N_NODES ints
  int* counts = deg + N_NODES;              // N_EDGES+1 ints (bincount length E+1)

  const int nz = N_NODES + N_EDGES + 1;
  wa_zero_i32<<<(nz + 255) / 256, 256, 0, stream>>>(deg, nz);
  wa_zero_accum<<<(N_NODES * EDGE_DIM + 255) / 256, 256, 0, stream>>>(out);
  int nblk = (N_TILES < SCATTER_BLOCKS) ? N_TILES : SCATTER_BLOCKS;
  wa_edge_scatter<<<nblk, 256, 0, stream>>>(e_lbl, edge_dst, out, deg, N_TILES);
  wa_bincount<<<(N_NODES + 255) / 256, 256, 0, stream>>>(deg, counts);
  wa_finalize<<<(N_NODES * OUT_DIM + 255) / 256, 256, 0, stream>>>(h, deg, counts, out);
}